// FastSpeech2ConformerLengthRegulator_17678085390937
// MI455X (gfx1250) — compile-verified
//
#include <hip/hip_runtime.h>
#include <stdint.h>

// FastSpeech2 Conformer length regulator for MI455X (gfx1250, wave32).
// Pure bandwidth problem (~110 MB of traffic -> ~4.7us at 23.3 TB/s).
// Uses CDNA5 async global<->LDS DMA (ASYNCcnt) for the row broadcast copies,
// and a wave32 shuffle scan (2 barriers total) for the duration cumsum.

#define BB 16
#define SS 512
#define DD 384
#define ROW_BYTES (DD * 4)          // 1536 B per token row
#define WAVES_PER_BLOCK 8

// ---------------------------------------------------------------------------
// Wave32 inclusive scan in registers (lowers to shuffles, no LDS traffic).
// ---------------------------------------------------------------------------
__device__ __forceinline__ int wave_incl_scan(int v, int lane) {
#pragma unroll
  for (int off = 1; off < 32; off <<= 1) {
    int n = __shfl_up(v, off, 32);
    if (lane >= off) v += n;
  }
  return v;
}

// ---------------------------------------------------------------------------
// Kernel 1: per-batch inclusive scan of durations (shuffle-based, 2 barriers),
// scatter frame->token map, emit lengths (int in ws, f32 into d_out tail).
// 512 threads = 16 waves per batch row.
// ---------------------------------------------------------------------------
__global__ void lr_scan_scatter(const int* __restrict__ dur,
                                int* __restrict__ idxmap,
                                int* __restrict__ lens,
                                float* __restrict__ out_len_f,
                                int L) {
  __shared__ int wsum[16];
  const int b    = blockIdx.x;
  const int t    = threadIdx.x;
  const int lane = t & 31;
  const int wv   = t >> 5;

  const int d = dur[b * SS + t];
  int s = wave_incl_scan(d, lane);          // inclusive scan within wave
  if (lane == 31) wsum[wv] = s;             // wave totals
  __syncthreads();

  if (wv == 0) {                            // scan the 16 wave totals
    int v = (lane < 16) ? wsum[lane] : 0;
    v = wave_incl_scan(v, lane);
    if (lane < 16) wsum[lane] = v;          // inclusive totals per wave
  }
  __syncthreads();

  const int base = (wv > 0) ? wsum[wv - 1] : 0;
  const int csum  = base + s;               // inclusive prefix over 512 tokens
  const int start = csum - d;               // exclusive prefix

  int* row = idxmap + (size_t)b * L;
  for (int k = start; k < csum; ++k) row[k] = t;   // disjoint ranges: race-free

  if (t == SS - 1) {
    lens[b]      = csum;                    // total expanded length
    out_len_f[b] = (float)csum;             // tuple output #2 (flattened as f32)
  }
}

// ---------------------------------------------------------------------------
// Kernel 2: one wave32 per output frame. Valid frames stream the 1536B token
// row through LDS with async DMA ops (data never touches VGPRs); padding
// frames store zeros. Branches are wave-uniform -> EXEC all-1s for async ops.
// ---------------------------------------------------------------------------
__global__ void lr_expand_copy(const float* __restrict__ hidden,
                               const int* __restrict__ idxmap,
                               const int* __restrict__ lens,
                               float* __restrict__ out,
                               int L) {
  __shared__ __align__(16) char smem[WAVES_PER_BLOCK * ROW_BYTES];  // 12 KB
  const int b    = blockIdx.y;
  const int wv   = threadIdx.x >> 5;
  const int lane = threadIdx.x & 31;
  const int j    = blockIdx.x * WAVES_PER_BLOCK + wv;     // wave-uniform
  if (j >= L) return;

  const int len = lens[b];                                // wave-uniform
  float* dst = out + ((size_t)b * L + j) * DD;

  if (j < len) {
    // ---- async DMA copy: global row -> LDS slot -> global output row ----
    const int tok = idxmap[(size_t)b * L + j];
    const float* src = hidden + ((size_t)b * SS + tok) * DD;

    // LDS byte offset of this wave's slot (low 32 bits of flat shared addr)
    uint32_t lds = (uint32_t)(uintptr_t)(&smem[0])
                 + (uint32_t)(wv * ROW_BYTES) + (uint32_t)(lane * 16);
    uint64_t gsrc = (uint64_t)(uintptr_t)src + (uint64_t)(lane * 16);
    uint64_t gdst = (uint64_t)(uintptr_t)dst + (uint64_t)(lane * 16);

    // 3 x 128b per lane = 512 B per wave per instruction; inst offset applies
    // to both the LDS and the global address (ISA 10.7 async addressing).
    asm volatile(
        "global_load_async_to_lds_b128 %0, %1, off\n\t"
        "global_load_async_to_lds_b128 %0, %1, off offset:512\n\t"
        "global_load_async_to_lds_b128 %0, %1, off offset:1024"
        :: "v"(lds), "v"(gsrc) : "memory");

    asm volatile("s_wait_asynccnt 0" ::: "memory");   // per-wave counter

    asm volatile(
        "global_store_async_from_lds_b128 %0, %1, off\n\t"
        "global_store_async_from_lds_b128 %0, %1, off offset:512\n\t"
        "global_store_async_from_lds_b128 %0, %1, off offset:1024"
        :: "v"(gdst), "v"(lds) : "memory");
    // Outstanding async stores drain at S_ENDPGM (implicit wait-idle).
  } else {
    // ---- zero padding region (harness poisons d_out, must overwrite) ----
    float4 z = make_float4(0.f, 0.f, 0.f, 0.f);
    float4* d4 = reinterpret_cast<float4*>(dst);
    d4[lane]      = z;
    d4[lane + 32] = z;
    d4[lane + 64] = z;
  }
}

// ---------------------------------------------------------------------------
// Host entry
// ---------------------------------------------------------------------------
extern "C" void kernel_launch(void* const* d_in, const int* in_sizes, int n_in,
                              void* d_out, int out_size, void* d_ws, size_t ws_size,
                              hipStream_t stream) {
  const float* hidden = (const float*)d_in[0];   // [B,S,D] f32
  const int*   dur    = (const int*)d_in[1];     // [B,S] i32
  float* out = (float*)d_out;                    // [B,L,D] f32 ++ [B] lengths

  // out_size = B*L*D + B  (tuple outputs flattened in return order)
  const int L = (out_size - BB) / (BB * DD);
  if (L <= 0) return;

  int* idxmap = (int*)d_ws;                                         // B*L ints
  int* lens   = (int*)((char*)d_ws + (size_t)BB * L * sizeof(int)); // B ints
  float* out_len_f = out + (size_t)BB * L * DD;

  lr_scan_scatter<<<dim3(BB), dim3(SS), 0, stream>>>(dur, idxmap, lens,
                                                     out_len_f, L);

  dim3 grid((L + WAVES_PER_BLOCK - 1) / WAVES_PER_BLOCK, BB);
  lr_expand_copy<<<grid, dim3(32 * WAVES_PER_BLOCK), 0, stream>>>(
      hidden, idxmap, lens, out, L);
}